// MultiGraphSage_11510512354048
// MI455X (gfx1250) — compile-verified
//
#include <hip/hip_runtime.h>
#include <hip/hip_bf16.h>
#include <math.h>

// ---------------------------------------------------------------------------
// MultiGraphSage on MI455X (gfx1250, wave32, WMMA)
//
//  layer: out = elu( (segment_mean(x[src]->dst)) @ W_l + b + x @ W_r )
//
//  Pipeline (all on `stream`):
//    1) zero agg[+deg]
//    2) scatter: atomic fadd of x[src] rows into agg[dst], deg counting
//    3) WMMA f32 16x16x4 GEMM epilogue-fused (mean, bias, ELU)
//  repeated for layer 2 with h.
// ---------------------------------------------------------------------------

#define NG 4
#define NN 50000
#define NE 800000
#define DD 64
#define TILES_PER_GRAPH (NN / 16)   // 3125, exact

typedef __attribute__((ext_vector_type(2))) float v2f;
typedef __attribute__((ext_vector_type(8))) float v8f;

// ---------------------------------------------------------------------------
__global__ void zero_f32_kernel(float* __restrict__ p, long long n) {
    long long i = (long long)blockIdx.x * blockDim.x + threadIdx.x;
    if (i < n) p[i] = 0.0f;
}

// ---------------------------------------------------------------------------
// 16 threads per edge; each thread handles 4 consecutive features (float4).
__global__ void sage_scatter_kernel(const float* __restrict__ feat,
                                    const long long* __restrict__ edge_index,
                                    float* __restrict__ agg,
                                    float* __restrict__ deg,
                                    int count_deg) {
    long long idx = (long long)blockIdx.x * blockDim.x + threadIdx.x;
    const long long total = (long long)NG * NE * 16;
    if (idx >= total) return;

    const int sub = (int)(idx & 15);          // which 4-float chunk
    const long long ge = idx >> 4;
    const int g = (int)(ge / NE);
    const int e = (int)(ge % NE);

    const long long* ebase = edge_index + (long long)g * 2 * NE;
    const int s = (int)ebase[e];              // src node
    const int d = (int)ebase[NE + e];         // dst node

    const float4 v = *(const float4*)(feat + ((long long)(g * NN + s)) * DD + sub * 4);
    float* ap = agg + ((long long)(g * NN + d)) * DD + sub * 4;

    __hip_atomic_fetch_add(ap + 0, v.x, __ATOMIC_RELAXED, __HIP_MEMORY_SCOPE_AGENT);
    __hip_atomic_fetch_add(ap + 1, v.y, __ATOMIC_RELAXED, __HIP_MEMORY_SCOPE_AGENT);
    __hip_atomic_fetch_add(ap + 2, v.z, __ATOMIC_RELAXED, __HIP_MEMORY_SCOPE_AGENT);
    __hip_atomic_fetch_add(ap + 3, v.w, __ATOMIC_RELAXED, __HIP_MEMORY_SCOPE_AGENT);

    if (count_deg && sub == 0) {
        __hip_atomic_fetch_add(deg + (long long)g * NN + d, 1.0f,
                               __ATOMIC_RELAXED, __HIP_MEMORY_SCOPE_AGENT);
    }
}

// ---------------------------------------------------------------------------
// One wave (32 lanes) computes a 16-node x 64-output tile:
//   out = elu( mean @ W_l + bias + in @ W_r ),  mean = agg * 1/max(deg,1)
//
// V_WMMA_F32_16X16X4_F32 operand layout (ISA 7.12.2):
//   A (16x4):  lane {half=lane>>4, m=lane&15} holds A[m][2*half], A[m][2*half+1]
//   B (4x16):  lane holds B[2*half][n], B[2*half+1][n], n = lane&15
//   C/D (16x16): element [M][N]: lane = N + 16*(M>=8), vgpr r = M % 8
__global__ void __launch_bounds__(256)
sage_gemm_elu_kernel(const float* __restrict__ in,
                     const float* __restrict__ agg,
                     const float* __restrict__ deg,
                     const float* __restrict__ Wl,
                     const float* __restrict__ Wr,
                     const float* __restrict__ bias,
                     float* __restrict__ out) {
    const int wave = (int)((blockIdx.x * blockDim.x + threadIdx.x) >> 5);
    const int lane = threadIdx.x & 31;
    if (wave >= NG * TILES_PER_GRAPH) return;   // wave-uniform: EXEC stays all-1s

    const int g = wave / TILES_PER_GRAPH;
    const int t = wave % TILES_PER_GRAPH;

    const int half = lane >> 4;
    const int m    = lane & 15;

    // ---- Load A fragments for this lane's row (mean and x), all K ----
    const long long row = (long long)g * NN + t * 16 + m;
    const float dg  = deg[row];
    const float inv = 1.0f / fmaxf(dg, 1.0f);

    const float* aggp = agg + row * DD;
    const float* inp  = in  + row * DD;

    v2f a_mean[16];
    v2f a_x[16];
#pragma unroll
    for (int kb = 0; kb < 16; ++kb) {
        const int k = kb * 4 + half * 2;
        v2f am = *(const v2f*)(aggp + k);
        am.x *= inv; am.y *= inv;
        a_mean[kb] = am;
        a_x[kb]    = *(const v2f*)(inp + k);
    }

    const float* wl = Wl   + (long long)g * DD * DD;
    const float* wr = Wr   + (long long)g * DD * DD;
    const float* bs = bias + (long long)g * DD;
    float* outp = out + ((long long)g * NN + t * 16) * DD;

    // ---- 4 column tiles of 16 outputs each ----
    for (int nt = 0; nt < 4; ++nt) {
        const int n0 = nt * 16 + m;           // this lane's output column
        const float bv = bs[n0];

        v8f c;
#pragma unroll
        for (int i = 0; i < 8; ++i) c[i] = bv;   // bias broadcast into every row

        // mean @ W_l
#pragma unroll
        for (int kb = 0; kb < 16; ++kb) {
            const int k = kb * 4 + half * 2;
            v2f b;
            b.x = wl[(k)     * DD + n0];
            b.y = wl[(k + 1) * DD + n0];
            c = __builtin_amdgcn_wmma_f32_16x16x4_f32(
                    false, a_mean[kb], false, b, (short)0, c, false, false);
        }
        // x @ W_r
#pragma unroll
        for (int kb = 0; kb < 16; ++kb) {
            const int k = kb * 4 + half * 2;
            v2f b;
            b.x = wr[(k)     * DD + n0];
            b.y = wr[(k + 1) * DD + n0];
            c = __builtin_amdgcn_wmma_f32_16x16x4_f32(
                    false, a_x[kb], false, b, (short)0, c, false, false);
        }

        // ELU epilogue + store: lane owns col n0, rows r + 8*half
#pragma unroll
        for (int r = 0; r < 8; ++r) {
            float v = c[r];
            v = (v > 0.0f) ? v : (__expf(v) - 1.0f);
            outp[(r + 8 * half) * DD + n0] = v;
        }
    }
}

// ---------------------------------------------------------------------------
extern "C" void kernel_launch(void* const* d_in, const int* in_sizes, int n_in,
                              void* d_out, int out_size, void* d_ws, size_t ws_size,
                              hipStream_t stream) {
    const float*     x    = (const float*)d_in[0];
    const long long* ei   = (const long long*)d_in[1];   // int64 per reference
    const float*     W_l1 = (const float*)d_in[2];
    const float*     W_r1 = (const float*)d_in[3];
    const float*     b1   = (const float*)d_in[4];
    const float*     W_l2 = (const float*)d_in[5];
    const float*     W_r2 = (const float*)d_in[6];
    const float*     b2   = (const float*)d_in[7];
    float* out = (float*)d_out;

    // Workspace layout: agg | deg | h   (~103.4 MB total)
    float* agg = (float*)d_ws;                               // NG*NN*DD
    float* deg = agg + (long long)NG * NN * DD;              // NG*NN
    float* h   = deg + (long long)NG * NN;                   // NG*NN*DD

    const long long agg_n = (long long)NG * NN * DD;
    const long long zer_n = agg_n + (long long)NG * NN;      // agg + deg
    const long long scat_threads = (long long)NG * NE * 16;

    const int ZB = 256;
    const int scat_blocks = (int)((scat_threads + ZB - 1) / ZB);
    const int gemm_waves  = NG * TILES_PER_GRAPH;            // 12500
    const int gemm_blocks = (gemm_waves + 7) / 8;            // 8 waves / 256-thread block

    // ---------------- Layer 1 ----------------
    zero_f32_kernel<<<(int)((zer_n + ZB - 1) / ZB), ZB, 0, stream>>>(agg, zer_n);
    sage_scatter_kernel<<<scat_blocks, ZB, 0, stream>>>(x, ei, agg, deg, 1);
    sage_gemm_elu_kernel<<<gemm_blocks, 256, 0, stream>>>(x, agg, deg, W_l1, W_r1, b1, h);

    // ---------------- Layer 2 ----------------
    zero_f32_kernel<<<(int)((agg_n + ZB - 1) / ZB), ZB, 0, stream>>>(agg, agg_n);
    sage_scatter_kernel<<<scat_blocks, ZB, 0, stream>>>(h, ei, agg, deg, 0);
    sage_gemm_elu_kernel<<<gemm_blocks, 256, 0, stream>>>(h, agg, deg, W_l2, W_r2, b2, out);
}